// GCN_E2_decline_leaky_slope_50087908606138
// MI455X (gfx1250) — compile-verified
//
#include <hip/hip_runtime.h>

// Problem constants (match the reference).
#define N_NODES 100000
#define N_EDGES 1600000
#define DIN     256
#define H1      128
#define H2      64
#define H3      32
#define DOUT    16
#define SLOPE   0.2f

typedef __attribute__((ext_vector_type(2))) float v2f;
typedef __attribute__((ext_vector_type(4))) float v4f;
typedef __attribute__((ext_vector_type(8))) float v8f;

__device__ __forceinline__ float lrelu(float v) {
    return v > 0.0f ? v : SLOPE * v;
}

// ---- CDNA5 async global->LDS copy (ASYNCcnt-tracked) -------------------
// Generic pointers: for __shared__, the low 32 bits are the LDS byte
// offset; for global, the generic address == the global address.
__device__ __forceinline__ void async_b128(void* lds, const void* g) {
    unsigned off = (unsigned)(unsigned long long)lds;
    asm volatile("global_load_async_to_lds_b128 %0, %1, off"
                 :: "v"(off), "v"(g) : "memory");
}
__device__ __forceinline__ void wait_async0() {
    asm volatile("s_wait_asynccnt 0x0" ::: "memory");
}

// =====================================================================
// Dense GEMM  C[nrows,NOUT] = A[nrows,K] @ B[K,NOUT]  via fp32 WMMA.
// A streams straight to registers (b64/lane/K-step, reused across all
// NT column tiles). B is staged into LDS in 32-row K-chunks with
// global_load_async_to_lds_b128, double-buffered so the copy of chunk
// c+1 overlaps the WMMA work on chunk c (s_wait_asynccnt + barrier).
// fp32 WMMA frags: A = {A[m][k0],A[m][k1]}, m=lane%16, k0=2*(lane/16);
//                  B = {B[k0][n],B[k0+1][n]}, n=lane%16;
//                  C/D: VGPR v -> M = v + 8*(lane/16), N = lane%16.
// =====================================================================
template<int K, int NOUT>
__global__ void __launch_bounds__(128)
k_gemm_wmma(const float* __restrict__ A, const float* __restrict__ Bg,
            float* __restrict__ C, int nrows)
{
    constexpr int NT  = NOUT / 16;
    constexpr int KC  = 32;                    // K rows per staged chunk
    constexpr int NCH = K / KC;
    constexpr int CF4 = KC * NOUT / 4;         // b128 copies per chunk
    __shared__ float Bs[2][KC * NOUT];

    const int tid  = threadIdx.x;
    const int lane = tid & 31;
    const int wave = tid >> 5;
    const int tiles = (nrows + 15) >> 4;
    int tile = blockIdx.x * 4 + wave;
    if (tile >= tiles) tile = tiles - 1;       // clamp; no divergent barrier
    const int m0 = tile << 4;
    const int ml = lane & 15;
    const int kh = (lane >> 4) << 1;           // 0 or 2
    int m = m0 + ml;
    if (m >= nrows) m = nrows - 1;             // clamp loads
    const float* __restrict__ arow = A + (size_t)m * K + kh;

    // Prologue: stage chunk 0.
#pragma unroll
    for (int i = tid; i < CF4; i += 128)
        async_b128(&Bs[0][i * 4], Bg + (size_t)i * 4);
    wait_async0();
    __syncthreads();

    v8f acc[NT];
#pragma unroll
    for (int t = 0; t < NT; ++t) { v8f z = {}; acc[t] = z; }

    for (int ch = 0; ch < NCH; ++ch) {
        const int cur = ch & 1;
        if (ch + 1 < NCH) {                    // prefetch next chunk
            const float* src = Bg + (size_t)(ch + 1) * KC * NOUT;
#pragma unroll
            for (int i = tid; i < CF4; i += 128)
                async_b128(&Bs[cur ^ 1][i * 4], src + (size_t)i * 4);
        }
        const int kbase = ch * KC;
#pragma unroll
        for (int kk = 0; kk < KC; kk += 4) {
            v2f a = *(const v2f*)(arow + kbase + kk);     // global_load_b64
            const float* lbp = &Bs[cur][(kk + kh) * NOUT + ml];
#pragma unroll
            for (int t = 0; t < NT; ++t) {
                v2f b;
                b.x = lbp[t * 16];                         // ds_load_b32
                b.y = lbp[t * 16 + NOUT];
                acc[t] = __builtin_amdgcn_wmma_f32_16x16x4_f32(
                    false, a, false, b, (short)0, acc[t], false, false);
            }
        }
        wait_async0();          // next chunk resident
        __syncthreads();        // everyone done reading cur before reuse
    }

    const int mrow = m0 + ((lane >> 4) << 3);
#pragma unroll
    for (int t = 0; t < NT; ++t) {
#pragma unroll
        for (int v = 0; v < 8; ++v) {
            const int mm = mrow + v;
            if (mm < nrows)
                C[(size_t)mm * NOUT + t * 16 + ml] = acc[t][v];
        }
    }
}

// =====================================================================
// Edge scatter:  out[row[e],:] += edge_w[e] * h[col[e],:]
// One wave per edge; lane covers F/32 consecutive floats (b128/b64
// gather out of L2, fp32 atomics into the L2-resident destination).
// =====================================================================
template<int F>
__global__ void __launch_bounds__(256)
k_spmm_atomic(const float* __restrict__ h, const int* __restrict__ row,
              const int* __restrict__ col, const float* __restrict__ ew,
              float* __restrict__ out)
{
    constexpr int V = F / 32;
    const int e = blockIdx.x * 8 + (threadIdx.x >> 5);
    if (e >= N_EDGES) return;
    const int lane = threadIdx.x & 31;
    const int r = row[e];
    const int c = col[e];
    const float w = ew[e];
    const float* __restrict__ src = h + (size_t)c * F + lane * V;
    float* dst = out + (size_t)r * F + lane * V;
    if constexpr (V == 4) {
        v4f t = *(const v4f*)src;                 // global_load_b128
        __hip_atomic_fetch_add(dst + 0, w * t.x, __ATOMIC_RELAXED, __HIP_MEMORY_SCOPE_AGENT);
        __hip_atomic_fetch_add(dst + 1, w * t.y, __ATOMIC_RELAXED, __HIP_MEMORY_SCOPE_AGENT);
        __hip_atomic_fetch_add(dst + 2, w * t.z, __ATOMIC_RELAXED, __HIP_MEMORY_SCOPE_AGENT);
        __hip_atomic_fetch_add(dst + 3, w * t.w, __ATOMIC_RELAXED, __HIP_MEMORY_SCOPE_AGENT);
    } else {
        v2f t = *(const v2f*)src;                 // global_load_b64
        __hip_atomic_fetch_add(dst + 0, w * t.x, __ATOMIC_RELAXED, __HIP_MEMORY_SCOPE_AGENT);
        __hip_atomic_fetch_add(dst + 1, w * t.y, __ATOMIC_RELAXED, __HIP_MEMORY_SCOPE_AGENT);
    }
}

// Initialize the SpMM accumulator with the broadcast bias (b128 stores).
template<int F>
__global__ void k_fill_bias(float* __restrict__ out, const float* __restrict__ b)
{
    const size_t i = (size_t)blockIdx.x * blockDim.x + threadIdx.x;
    if (i < (size_t)N_NODES * F / 4) {
        const size_t j = i * 4;
        v4f bv = *(const v4f*)&b[j & (F - 1)];
        *(v4f*)&out[j] = bv;
    }
}

// h = lrelu(h) * drop_mask   (elementwise over N x H1, b128 accesses)
__global__ void k_act_drop(float* __restrict__ h, const float* __restrict__ mask)
{
    const size_t i = (size_t)blockIdx.x * blockDim.x + threadIdx.x;
    if (i < (size_t)N_NODES * H1 / 4) {
        v4f v = *(const v4f*)&h[i * 4];
        v4f mk = *(const v4f*)&mask[i * 4];
        v.x = lrelu(v.x) * mk.x;
        v.y = lrelu(v.y) * mk.y;
        v.z = lrelu(v.z) * mk.z;
        v.w = lrelu(v.w) * mk.w;
        *(v4f*)&h[i * 4] = v;
    }
}

// =====================================================================
// Head: per 16-node tile,
//   t = lrelu( lrelu(hB) @ W3 + b3 )   (16x64 @ 64x32)
//   out = t @ W4 + b4                  (16x32 @ 32x16)
// The intermediate tile is staged through LDS to convert the WMMA
// C/D layout into the A-fragment layout for the second WMMA chain.
// =====================================================================
__global__ void __launch_bounds__(128)
k_head_wmma(const float* __restrict__ hB, const float* __restrict__ W3,
            const float* __restrict__ b3, const float* __restrict__ W4,
            const float* __restrict__ b4, float* __restrict__ out)
{
    __shared__ float Tlds[4][16 * 32];
    const int lane = threadIdx.x & 31;
    const int wave = threadIdx.x >> 5;
    const int tiles = (N_NODES + 15) >> 4;
    int tile = blockIdx.x * 4 + wave;
    if (tile >= tiles) tile = tiles - 1;          // no divergent barrier
    const int m0 = tile << 4;
    const int ml = lane & 15;
    const int kh = (lane >> 4) << 1;
    int m = m0 + ml;
    if (m >= N_NODES) m = N_NODES - 1;

    // ---- layer 3: lrelu(hB) @ W3 ----
    const float* __restrict__ arow = hB + (size_t)m * H2 + kh;
    v8f acc3[2];
    { v8f z = {}; acc3[0] = z; acc3[1] = z; }
#pragma unroll
    for (int k4 = 0; k4 < H2; k4 += 4) {
        v2f a = *(const v2f*)(arow + k4);
        a.x = lrelu(a.x);
        a.y = lrelu(a.y);
        const float* bp = W3 + (size_t)(k4 + kh) * H3 + ml;
#pragma unroll
        for (int t = 0; t < 2; ++t) {
            v2f b;
            b.x = bp[t * 16];
            b.y = bp[t * 16 + H3];
            acc3[t] = __builtin_amdgcn_wmma_f32_16x16x4_f32(
                false, a, false, b, (short)0, acc3[t], false, false);
        }
    }
    // T = lrelu(acc3 + b3) -> LDS, row-major [16][32]
    {
        const int mr = (lane >> 4) << 3;
#pragma unroll
        for (int t = 0; t < 2; ++t) {
            const float bb = b3[t * 16 + ml];
#pragma unroll
            for (int v = 0; v < 8; ++v)
                Tlds[wave][(mr + v) * 32 + t * 16 + ml] = lrelu(acc3[t][v] + bb);
        }
    }
    __syncthreads();

    // ---- layer 4: T @ W4 ----
    v8f acc4 = {};
#pragma unroll
    for (int k4 = 0; k4 < H3; k4 += 4) {
        v2f a = *(const v2f*)&Tlds[wave][ml * 32 + k4 + kh];   // ds_load_b64
        const float* bp = W4 + (size_t)(k4 + kh) * DOUT + ml;
        v2f b;
        b.x = bp[0];
        b.y = bp[DOUT];
        acc4 = __builtin_amdgcn_wmma_f32_16x16x4_f32(
            false, a, false, b, (short)0, acc4, false, false);
    }
    const int mrow = m0 + ((lane >> 4) << 3);
    const float bb = b4[ml];
#pragma unroll
    for (int v = 0; v < 8; ++v) {
        const int mm = mrow + v;
        if (mm < N_NODES) out[(size_t)mm * DOUT + ml] = acc4[v] + bb;
    }
}

// =====================================================================
extern "C" void kernel_launch(void* const* d_in, const int* in_sizes, int n_in,
                              void* d_out, int out_size, void* d_ws, size_t ws_size,
                              hipStream_t stream) {
    (void)in_sizes; (void)n_in; (void)out_size; (void)ws_size;
    const float* x    = (const float*)d_in[0];
    const int*   row  = (const int*)  d_in[1];
    const int*   col  = (const int*)  d_in[2];
    const float* ew   = (const float*)d_in[3];
    const float* W1   = (const float*)d_in[4];
    const float* b1   = (const float*)d_in[5];
    const float* W2   = (const float*)d_in[6];
    const float* b2   = (const float*)d_in[7];
    const float* W3   = (const float*)d_in[8];
    const float* b3   = (const float*)d_in[9];
    const float* W4   = (const float*)d_in[10];
    const float* b4   = (const float*)d_in[11];
    const float* mask = (const float*)d_in[12];
    float* out = (float*)d_out;

    // Workspace: two 51.2MB ping-pong buffers.
    float* buf0 = (float*)d_ws;                       // h0, later h2
    float* buf1 = buf0 + (size_t)N_NODES * H1;        // hA, later hB

    constexpr int TILES      = (N_NODES + 15) / 16;   // 6250
    constexpr int GEMM_BLKS  = (TILES + 3) / 4;       // 4 waves/block
    constexpr int EDGE_BLKS  = (N_EDGES + 7) / 8;     // 8 edges/block
    constexpr int EV1        = ((N_NODES * H1 / 4) + 255) / 256;
    constexpr int EV2        = ((N_NODES * H2 / 4) + 255) / 256;

    // h0 = x @ W1
    k_gemm_wmma<DIN, H1><<<GEMM_BLKS, 128, 0, stream>>>(x, W1, buf0, N_NODES);
    // hA = b1; hA += scatter(edge_w * h0[col])
    k_fill_bias<H1><<<EV1, 256, 0, stream>>>(buf1, b1);
    k_spmm_atomic<H1><<<EDGE_BLKS, 256, 0, stream>>>(buf0, row, col, ew, buf1);
    // hA = lrelu(hA) * drop_mask
    k_act_drop<<<EV1, 256, 0, stream>>>(buf1, mask);
    // h2 = hA @ W2   (reuse buf0)
    k_gemm_wmma<H1, H2><<<GEMM_BLKS, 128, 0, stream>>>(buf1, W2, buf0, N_NODES);
    // hB = b2; hB += scatter(edge_w * h2[col])   (reuse buf1)
    k_fill_bias<H2><<<EV2, 256, 0, stream>>>(buf1, b2);
    k_spmm_atomic<H2><<<EDGE_BLKS, 256, 0, stream>>>(buf0, row, col, ew, buf1);
    // out = lrelu(lrelu(hB) @ W3 + b3) @ W4 + b4
    k_head_wmma<<<GEMM_BLKS, 128, 0, stream>>>(buf1, W3, b3, W4, b4, out);
}